// Model_15307263443707
// MI455X (gfx1250) — compile-verified
//
#include <hip/hip_runtime.h>
#include <stdint.h>

typedef unsigned int u32;

#define TPB 128
#define NBLK_PER_THREAD 4

// AES S-box (matches reference _SBOX_NP)
__device__ __constant__ unsigned char c_SBOX[256] = {
  99,124,119,123,242,107,111,197, 48,  1,103, 43,254,215,171,118,
 202,130,201,125,250, 89, 71,240,173,212,162,175,156,164,114,192,
 183,253,147, 38, 54, 63,247,204, 52,165,229,241,113,216, 49, 21,
   4,199, 35,195, 24,150,  5,154,  7, 18,128,226,235, 39,178,117,
   9,131, 44, 26, 27,110, 90,160, 82, 59,214,179, 41,227, 47,132,
  83,209,  0,237, 32,252,177, 91,106,203,190, 57, 74, 76, 88,207,
 208,239,170,251, 67, 77, 51,133, 69,249,  2,127, 80, 60,159,168,
  81,163, 64,143,146,157, 56,245,188,182,218, 33, 16,255,243,210,
 205, 12, 19,236, 95,151, 68, 23,196,167,126, 61,100, 93, 25,115,
  96,129, 79,220, 34, 42,144,136, 70,238,184, 20,222, 94, 11,219,
 224, 50, 58, 10, 73,  6, 36, 92,194,211,172, 98,145,149,228,121,
 231,200, 55,109,141,213, 78,169,108, 86,244,234,101,122,174,  8,
 186,120, 37, 46, 28,166,180,198,232,221,116, 31, 75,189,139,138,
 112, 62,181,102, 72,  3,246, 14, 97, 53, 87,185,134,193, 29,158,
 225,248,152, 17,105,217,142,148,155, 30,135,233,206, 85, 40,223,
 140,161,137, 13,191,230, 66,104, 65,153, 45, 15,176, 84,187, 22
};

__device__ __forceinline__ u32 rotl8 (u32 v) { return (v << 8)  | (v >> 24); }
__device__ __forceinline__ u32 rotl16(u32 v) { return (v << 16) | (v >> 16); }
__device__ __forceinline__ u32 rotl24(u32 v) { return (v << 24) | (v >> 8);  }

// T0[x] = { 2*S, S, S, 3*S } little-endian (state row r at bits [8r,8r+8)).
// Ti = rotl(T0, 8*i).   S[x] = (T0[x] >> 8) & 0xff.

#define BYTE0(v) ((v) & 0xffu)
#define BYTE1(v) (((v) >> 8) & 0xffu)
#define BYTE2(v) (((v) >> 16) & 0xffu)
#define BYTE3(v) ((v) >> 24)

// per-lane replica: entry x for lane L at dword x*32 + L -> banks {L, L+32}, conflict-free
#define TLU(x) tp[((x) << 5)]

#define ENC_COL(a,b,c,d,k) ( TLU(BYTE0(a)) ^ rotl8(TLU(BYTE1(b))) ^ \
                             rotl16(TLU(BYTE2(c))) ^ rotl24(TLU(BYTE3(d))) ^ (k) )

#define FIN_COL(a,b,c,d,k) ( ( ((TLU(BYTE0(a)) >> 8) & 0x000000ffu)   \
                             | ( TLU(BYTE1(b))       & 0x0000ff00u)   \
                             | ((TLU(BYTE2(c)) << 8) & 0x00ff0000u)   \
                             | ((TLU(BYTE3(d)) << 16)& 0xff000000u) ) ^ (k) )

// Issue 4 async global->LDS b128 loads for one 64B AES block.
// NOTE: the ISA adds the instruction offset to BOTH the global and the LDS
// address; the ldsA[] base registers are pre-compensated by -q*16.
#define STAGE(par, gptr)                                                           \
    do {                                                                           \
        asm volatile("global_load_async_to_lds_b128 %0, %1, off"                   \
                     :: "v"(ldsA[par][0]), "v"(gptr) : "memory");                  \
        asm volatile("global_load_async_to_lds_b128 %0, %1, off offset:16"         \
                     :: "v"(ldsA[par][1]), "v"(gptr) : "memory");                  \
        asm volatile("global_load_async_to_lds_b128 %0, %1, off offset:32"         \
                     :: "v"(ldsA[par][2]), "v"(gptr) : "memory");                  \
        asm volatile("global_load_async_to_lds_b128 %0, %1, off offset:48"         \
                     :: "v"(ldsA[par][3]), "v"(gptr) : "memory");                  \
    } while (0)

__global__ __launch_bounds__(TPB)
void aes128_ecb_kernel(const int* __restrict__ pt, const int* __restrict__ keyb,
                       int* __restrict__ out, int nblocks)
{
    __shared__ u32 sT[256 * 32];       // 32 KB: T0 replicated per lane
    __shared__ u32 sIn[2 * 2048];      // 16 KB: double-buffered plaintext staging
    __shared__ u32 sRK[44];            // round keys (column words)

    const int tid = threadIdx.x;

    // ---- build bank-exclusive replicated T0 table ----
    for (int idx = tid; idx < 256 * 32; idx += TPB) {
        u32 s  = (u32)c_SBOX[idx >> 5];
        u32 s2 = ((s << 1) ^ (((s >> 7) & 1u) * 0x1bu)) & 0xffu;  // xtime
        u32 s3 = s2 ^ s;
        sT[idx] = s2 | (s << 8) | (s << 16) | (s3 << 24);
    }

    // ---- key schedule (one thread per workgroup; cheap, amortized) ----
    if (tid == 0) {
        u32 w[44];
        #pragma unroll
        for (int j = 0; j < 4; ++j)
            w[j] = (u32)(keyb[4*j] & 0xff)
                 | ((u32)(keyb[4*j+1] & 0xff) << 8)
                 | ((u32)(keyb[4*j+2] & 0xff) << 16)
                 | ((u32)(keyb[4*j+3] & 0xff) << 24);
        const u32 rcon[10] = {1,2,4,8,16,32,64,128,27,54};
        #pragma unroll
        for (int i = 4; i < 44; ++i) {
            u32 t = w[i-1];
            if ((i & 3) == 0) {
                u32 r = (t >> 8) | (t << 24);            // RotWord
                t = (u32)c_SBOX[r & 0xff]
                  | ((u32)c_SBOX[(r >> 8)  & 0xff] << 8)
                  | ((u32)c_SBOX[(r >> 16) & 0xff] << 16)
                  | ((u32)c_SBOX[(r >> 24)       ] << 24);
                t ^= rcon[(i >> 2) - 1];
            }
            w[i] = w[i-4] ^ t;
        }
        #pragma unroll
        for (int i = 0; i < 44; ++i) sRK[i] = w[i];
    }

    __syncthreads();

    // ---- broadcast round keys into SGPRs ----
    u32 rk[44];
    #pragma unroll
    for (int i = 0; i < 44; ++i)
        rk[i] = (u32)__builtin_amdgcn_readfirstlane((int)sRK[i]);

    const int  lane = tid & 31;              // wave32
    const u32* tp   = sT + lane;             // this lane's T-table replica

    // LDS byte offsets for async staging: buffer p, chunk q of thread t lives at
    // sIn + p*8192 + q*2048 + t*16 bytes (chunk-major => 2-pass-minimum ds_load_b128).
    // Pre-compensate -q*16 for the shared instruction offset (see STAGE note).
    const u32 sInBase = (u32)(uintptr_t)(&sIn[0]);
    u32 ldsA[2][4];
    #pragma unroll
    for (int p = 0; p < 2; ++p)
        #pragma unroll
        for (int q = 0; q < 4; ++q)
            ldsA[p][q] = sInBase + (u32)p * 8192u + (u32)q * 2048u
                       + (u32)tid * 16u - (u32)q * 16u;

    const int gsz = (int)(gridDim.x * blockDim.x);
    int b = (int)(blockIdx.x * blockDim.x) + tid;

    const int* gp = pt + (size_t)b * 16;
    if (b < nblocks) STAGE(0, gp);           // preload first block (ASYNCcnt += 4)

    int parity = 0;
    for (; b < nblocks; b += gsz) {
        const int* gpn = gp + (size_t)gsz * 16;

        if (b + gsz < nblocks) {
            STAGE(parity ^ 1, gpn);          // overlap next fetch with compute
            // in-order completion: <=4 outstanding => current buffer's 4 are done
            asm volatile("s_wait_asynccnt 0x4" ::: "memory");
        } else {
            asm volatile("s_wait_asynccnt 0x0" ::: "memory");
        }

        const u32* sp = sIn + (parity << 11) + (tid << 2);
        uint4 q0 = *(const uint4*)(sp);
        uint4 q1 = *(const uint4*)(sp + 512);
        uint4 q2 = *(const uint4*)(sp + 1024);
        uint4 q3 = *(const uint4*)(sp + 1536);

        u32 c0 = (q0.x & 0xffu) | ((q0.y & 0xffu) << 8) | ((q0.z & 0xffu) << 16) | ((q0.w & 0xffu) << 24);
        u32 c1 = (q1.x & 0xffu) | ((q1.y & 0xffu) << 8) | ((q1.z & 0xffu) << 16) | ((q1.w & 0xffu) << 24);
        u32 c2 = (q2.x & 0xffu) | ((q2.y & 0xffu) << 8) | ((q2.z & 0xffu) << 16) | ((q2.w & 0xffu) << 24);
        u32 c3 = (q3.x & 0xffu) | ((q3.y & 0xffu) << 8) | ((q3.z & 0xffu) << 16) | ((q3.w & 0xffu) << 24);

        c0 ^= rk[0]; c1 ^= rk[1]; c2 ^= rk[2]; c3 ^= rk[3];

        #pragma unroll
        for (int r = 1; r <= 9; ++r) {
            u32 t0 = ENC_COL(c0, c1, c2, c3, rk[4*r + 0]);
            u32 t1 = ENC_COL(c1, c2, c3, c0, rk[4*r + 1]);
            u32 t2 = ENC_COL(c2, c3, c0, c1, rk[4*r + 2]);
            u32 t3 = ENC_COL(c3, c0, c1, c2, rk[4*r + 3]);
            c0 = t0; c1 = t1; c2 = t2; c3 = t3;
        }

        u32 d0 = FIN_COL(c0, c1, c2, c3, rk[40]);
        u32 d1 = FIN_COL(c1, c2, c3, c0, rk[41]);
        u32 d2 = FIN_COL(c2, c3, c0, c1, rk[42]);
        u32 d3 = FIN_COL(c3, c0, c1, c2, rk[43]);

        int4* po = (int4*)(out + (size_t)b * 16);
        po[0] = make_int4((int)(d0 & 0xff), (int)((d0 >> 8) & 0xff), (int)((d0 >> 16) & 0xff), (int)(d0 >> 24));
        po[1] = make_int4((int)(d1 & 0xff), (int)((d1 >> 8) & 0xff), (int)((d1 >> 16) & 0xff), (int)(d1 >> 24));
        po[2] = make_int4((int)(d2 & 0xff), (int)((d2 >> 8) & 0xff), (int)((d2 >> 16) & 0xff), (int)(d2 >> 24));
        po[3] = make_int4((int)(d3 & 0xff), (int)((d3 >> 8) & 0xff), (int)((d3 >> 16) & 0xff), (int)(d3 >> 24));

        gp = gpn;
        parity ^= 1;
    }
}

extern "C" void kernel_launch(void* const* d_in, const int* in_sizes, int n_in,
                              void* d_out, int out_size, void* d_ws, size_t ws_size,
                              hipStream_t stream)
{
    const int* pt  = (const int*)d_in[0];   // plaintext, (B,16) int32 bytes
    const int* key = (const int*)d_in[1];   // key, (16,) int32 bytes
    int* out = (int*)d_out;                 // int32 output, same layout as reference

    const int nblocks = in_sizes[0] / 16;   // B = 2^21
    const int work_per_wg = TPB * NBLK_PER_THREAD;
    int grid = (nblocks + work_per_wg - 1) / work_per_wg;
    if (grid < 1) grid = 1;

    aes128_ecb_kernel<<<grid, TPB, 0, stream>>>(pt, key, out, nblocks);
}